// LightGCN_56538949484988
// MI455X (gfx1250) — compile-verified
//
#include <hip/hip_runtime.h>

// LightGCN on MI455X (gfx1250, wave32).
// - 3x SpMM via per-edge wave32 gather + hardware f32 L2 atomics (L2-resident working set).
// - Prediction head with V_WMMA_F32_16X16X4_F32 (exact fp32 matrix path), LDS-staged A tiles,
//   fused bias + leaky_relu(0.2) + w2 dot-product epilogue (shuffle-reduce + atomic).

typedef float v2f __attribute__((ext_vector_type(2)));
typedef float v8f __attribute__((ext_vector_type(8)));

#define C_DIM 128

__device__ __forceinline__ void atomAddF(float* p, float v) {
  unsafeAtomicAdd(p, v);   // lowers to global_atomic_add_f32 on gfx1250
}

// final = emb ; hA = 0 ; deg = 0
__global__ void k_init(const float* __restrict__ emb, float* __restrict__ fin,
                       float* __restrict__ hA, float* __restrict__ deg,
                       int NC, int n_nodes) {
  int i = blockIdx.x * blockDim.x + threadIdx.x;
  if (i < NC) { fin[i] = emb[i]; hA[i] = 0.0f; }
  if (i < n_nodes) deg[i] = 0.0f;
}

__global__ void k_deg(const int* __restrict__ row, float* __restrict__ deg, int E) {
  int e = blockIdx.x * blockDim.x + threadIdx.x;
  if (e < E) atomAddF(&deg[row[e]], 1.0f);
}

__global__ void k_dinv(float* __restrict__ deg, int n) {
  int i = blockIdx.x * blockDim.x + threadIdx.x;
  if (i < n) {
    float d = deg[i];
    deg[i] = (d > 0.0f) ? rsqrtf(fmaxf(d, 1e-12f)) : 0.0f;
  }
}

__global__ void k_norm(const int* __restrict__ row, const int* __restrict__ col,
                       const float* __restrict__ dinv, float* __restrict__ norm, int E) {
  int e = blockIdx.x * blockDim.x + threadIdx.x;
  if (e < E) norm[e] = dinv[row[e]] * dinv[col[e]];
}

// One wave32 per edge: 4 channels/lane, float4 gather, 4 HW f32 atomics scatter.
__global__ void k_spmm(const int* __restrict__ row, const int* __restrict__ col,
                       const float* __restrict__ norm, const float* __restrict__ hin,
                       float* __restrict__ hout, int E) {
  int e = blockIdx.x * (blockDim.x >> 5) + (threadIdx.x >> 5);
  if (e >= E) return;
  int lane = threadIdx.x & 31;
  float w = norm[e];
  const float4 v = *(const float4*)(hin + (size_t)col[e] * C_DIM + lane * 4);
  float* dst = hout + (size_t)row[e] * C_DIM + lane * 4;
  atomAddF(dst + 0, w * v.x);
  atomAddF(dst + 1, w * v.y);
  atomAddF(dst + 2, w * v.z);
  atomAddF(dst + 3, w * v.w);
}

// fin = (fin + h) * scale ; optionally zero the next scatter target
__global__ void k_accum(float* __restrict__ fin, const float* __restrict__ h,
                        float* __restrict__ zbuf, float scale, int NC) {
  int i = blockIdx.x * blockDim.x + threadIdx.x;
  if (i < NC) {
    fin[i] = (fin[i] + h[i]) * scale;
    if (zbuf) zbuf[i] = 0.0f;
  }
}

// X[r] = concat(final[idL], final[idR]); one wave per row, float4 copies.
__global__ void k_buildx(const float* __restrict__ fin, const int* __restrict__ n_id,
                         float* __restrict__ X, int b, int rows, int neg) {
  int r = blockIdx.x * (blockDim.x >> 5) + (threadIdx.x >> 5);
  if (r >= rows) return;
  int lane = threadIdx.x & 31;
  int idL, idR;
  if (r < b) { idL = n_id[r];           idR = n_id[b + r]; }
  else       { int j = r - b; idL = n_id[j / neg]; idR = n_id[2 * b + j]; }
  const float4* a  = (const float4*)(fin + (size_t)idL * C_DIM);
  const float4* bb = (const float4*)(fin + (size_t)idR * C_DIM);
  float4* dst = (float4*)(X + (size_t)r * (2 * C_DIM));
  dst[lane]      = a[lane];    // 32 float4 = first 128 floats
  dst[32 + lane] = bb[lane];   // second 128 floats
}

__global__ void k_initout(float* __restrict__ out, const float* __restrict__ b2, int n) {
  int i = blockIdx.x * blockDim.x + threadIdx.x;
  if (i < n) out[i] = b2[0];
}

// Fused head: one block per 16 M-rows; 8 waves cover N=128 in 16-wide strips.
// Each wave: 64x V_WMMA_F32_16X16X4_F32 over K=256, then bias+leakyReLU+dot(w2)
// with 16-lane shuffle reduction; accumulate into out via HW atomics.
__global__ void k_head(const float* __restrict__ X, const float* __restrict__ W1,
                       const float* __restrict__ b1, const float* __restrict__ w2,
                       float* __restrict__ out) {
  __shared__ float sX[16 * 2 * C_DIM];            // 16 rows x 256 cols = 16 KB
  const int row0 = blockIdx.x * 16;
  const int tid  = threadIdx.x;
  const int wave = tid >> 5;
  const int lane = tid & 31;

  // cooperative coalesced stage of the 16x256 X tile
  const float4* src = (const float4*)(X + (size_t)row0 * (2 * C_DIM));
  float4* s4 = (float4*)sX;
  for (int i = tid; i < (16 * 256) / 4; i += 256) s4[i] = src[i];
  __syncthreads();

  const int n0   = wave * 16;
  const int r    = lane & 15;      // M row (A) / N col (B,C)
  const int ksel = lane >> 4;      // 0: K pair {0,1}; 1: K pair {2,3}
  v8f acc = {};
  for (int k0 = 0; k0 < 2 * C_DIM; k0 += 4) {
    const int ka = k0 + 2 * ksel;
    // A 16x4 f32: lanes 0-15 rows M=0..15 with K=k0,k0+1; lanes 16-31 K=k0+2,k0+3
    v2f a = *(const v2f*)(sX + r * (2 * C_DIM) + ka);
    // B 4x16 f32: row-striped across lanes; lane half selects K rows {k0,k0+1}/{k0+2,k0+3}
    v2f bv;
    bv.x = W1[ka * C_DIM + n0 + r];
    bv.y = W1[(ka + 1) * C_DIM + n0 + r];
    acc = __builtin_amdgcn_wmma_f32_16x16x4_f32(
        /*neg_a=*/false, a, /*neg_b=*/false, bv,
        /*c_mod=*/(short)0, acc, /*reuse_a=*/false, /*reuse_b=*/false);
  }

  // C/D layout: VGPR j -> M=row0+j (lanes 0-15, N=n0+lane) and M=row0+8+j (lanes 16-31)
  const float bias = b1[n0 + r];
  const float wv   = w2[n0 + r];
#pragma unroll
  for (int j = 0; j < 8; ++j) {
    float z = acc[j] + bias;
    z = (z > 0.0f) ? z : 0.2f * z;                // leaky_relu slope 0.2
    float p = z * wv;
    p += __shfl_xor(p, 1, 32);
    p += __shfl_xor(p, 2, 32);
    p += __shfl_xor(p, 4, 32);
    p += __shfl_xor(p, 8, 32);                    // full sum within each 16-lane half
    if (r == 0) atomAddF(&out[row0 + 8 * ksel + j], p);
  }
}

extern "C" void kernel_launch(void* const* d_in, const int* in_sizes, int n_in,
                              void* d_out, int out_size, void* d_ws, size_t ws_size,
                              hipStream_t stream) {
  (void)n_in; (void)out_size; (void)ws_size;
  const int*   edge = (const int*)d_in[1];       // [2,E] row-major: row then col
  const int*   n_id = (const int*)d_in[3];
  const float* emb  = (const float*)d_in[5];
  const float* w1   = (const float*)d_in[6];
  const float* b1   = (const float*)d_in[7];
  const float* w2   = (const float*)d_in[8];
  const float* b2   = (const float*)d_in[9];

  const int E       = in_sizes[1] / 2;
  const int nid     = in_sizes[3];
  const int n_nodes = in_sizes[5] / C_DIM;
  const int NEG     = 5;                          // reference constant
  const int b       = nid / (2 + NEG);
  const int rows    = b + NEG * b;                // 24576, multiple of 16
  const int NC      = n_nodes * C_DIM;

  const int* rowi = edge;
  const int* coli = edge + E;

  // workspace layout (floats), 16B-aligned segments
  float* ws   = (float*)d_ws;
  const int padN = ((n_nodes + 63) / 64) * 64;
  const int padE = ((E + 63) / 64) * 64;
  float* deg  = ws;                 // -> dinv in place
  float* norm = deg + padN;
  float* hA   = norm + padE;
  float* hB   = hA + NC;
  float* fin  = hB + NC;
  float* X    = hA;                 // reuse hA after propagation (rows*256 <= NC)
  float* out  = (float*)d_out;

  const int T   = 256;
  const int WPB = T / 32;
  const int spmmBlocks = (E + WPB - 1) / WPB;

  k_init<<<(NC + T - 1) / T, T, 0, stream>>>(emb, fin, hA, deg, NC, n_nodes);
  k_deg <<<(E + T - 1) / T, T, 0, stream>>>(rowi, deg, E);
  k_dinv<<<(n_nodes + T - 1) / T, T, 0, stream>>>(deg, n_nodes);
  k_norm<<<(E + T - 1) / T, T, 0, stream>>>(rowi, coli, deg, norm, E);

  // layer 1: emb -> hA ; fin += hA ; zero hB
  k_spmm <<<spmmBlocks, T, 0, stream>>>(rowi, coli, norm, emb, hA, E);
  k_accum<<<(NC + T - 1) / T, T, 0, stream>>>(fin, hA, hB, 1.0f, NC);
  // layer 2: hA -> hB ; fin += hB ; zero hA
  k_spmm <<<spmmBlocks, T, 0, stream>>>(rowi, coli, norm, hA, hB, E);
  k_accum<<<(NC + T - 1) / T, T, 0, stream>>>(fin, hB, hA, 1.0f, NC);
  // layer 3: hB -> hA ; fin = (fin + hA) / 4
  k_spmm <<<spmmBlocks, T, 0, stream>>>(rowi, coli, norm, hB, hA, E);
  k_accum<<<(NC + T - 1) / T, T, 0, stream>>>(fin, hA, nullptr, 0.25f, NC);

  // gather + fused WMMA head
  k_buildx <<<(rows + WPB - 1) / WPB, T, 0, stream>>>(fin, n_id, X, b, rows, NEG);
  k_initout<<<(rows + T - 1) / T, T, 0, stream>>>(out, b2, rows);
  k_head   <<<rows / 16, T, 0, stream>>>(X, w1, b1, w2, out);
}